// Qwen2Attention_61186104099190
// MI455X (gfx1250) — compile-verified
//
#include <hip/hip_runtime.h>
#include <math.h>

typedef float v2f __attribute__((ext_vector_type(2)));
typedef float v8f __attribute__((ext_vector_type(8)));

constexpr int B    = 8;
constexpr int QL   = 4;
constexpr int PAST = 4096;
constexpr int HID  = 3584;
constexpr int NH   = 28;
constexpr int NKV  = 4;
constexpr int HD   = 128;
constexpr int KVL  = PAST + QL;      // 4100
constexpr int NREP = NH / NKV;       // 7
constexpr int ROWS = B * QL;         // 32
constexpr int NQ   = NH * HD;        // 3584
constexpr int NK   = NKV * HD;       // 512
constexpr int NTOT = NQ + 2 * NK;    // 4608
constexpr float SCALE = 0.0883883476483184405f;  // 1/sqrt(128)

// ---------------------------------------------------------------------------
// Kernel 1: fused QKV projection via fp32 WMMA.  C(16x16) = X(16xK) * W(Kx16)
// X = hidden flattened [32][3584].  Output qkv[32][4608] (bias included).
// A frag (16x4 f32): lanes0-15 row=l, K={k0,k0+1}; lanes16-31 row=l, K={k0+2,k0+3}
// B frag (4x16 f32): lanes0-15 col=l, K={k0,k0+1}; lanes16-31 col=l, K={k0+2,k0+3}
// C frag: vgpr r -> lanes0-15 (M=r,N=l), lanes16-31 (M=8+r,N=l)
// ---------------------------------------------------------------------------
__global__ __launch_bounds__(32)
void k_qkv_gemm(const float* __restrict__ X,
                const float* __restrict__ Wq, const float* __restrict__ bq,
                const float* __restrict__ Wk, const float* __restrict__ bk,
                const float* __restrict__ Wv, const float* __restrict__ bv,
                float* __restrict__ qkv)
{
    const int lane   = threadIdx.x;
    const int n0     = blockIdx.x * 16;     // global output column tile
    const int tile_m = blockIdx.y;          // 0..1

    const float* W; const float* bias; int wld, ncol;
    if (n0 < NQ)           { W = Wq; bias = bq; ncol = n0;           wld = NQ; }
    else if (n0 < NQ + NK) { W = Wk; bias = bk; ncol = n0 - NQ;      wld = NK; }
    else                   { W = Wv; bias = bv; ncol = n0 - NQ - NK; wld = NK; }

    const int half = lane >> 4;
    const int l    = lane & 15;
    const float* Arow = X + (size_t)(tile_m * 16 + l) * HID;
    const float* Bcol = W + (ncol + l);

    v8f c = {};
    for (int k0 = 0; k0 < HID; k0 += 4) {
        const int ka = k0 + half * 2;
        v2f a, b;
        a.x = Arow[ka];
        a.y = Arow[ka + 1];
        b.x = Bcol[(size_t)ka * wld];
        b.y = Bcol[(size_t)(ka + 1) * wld];
        c = __builtin_amdgcn_wmma_f32_16x16x4_f32(false, a, false, b,
                                                  (short)0, c, false, false);
    }

    const float bb = bias[ncol + l];
    const int mb = tile_m * 16 + half * 8;
#pragma unroll
    for (int r = 0; r < 8; ++r)
        qkv[(size_t)(mb + r) * NTOT + n0 + l] = c[r] + bb;
}

// ---------------------------------------------------------------------------
// Kernel 2: RoPE on Q and new K, scatter to head-major layouts.
// q_rope[(b*NH+h)*QL + s][HD], k_new[(b*NKV+kh)*QL + s][HD]
// ---------------------------------------------------------------------------
__global__ __launch_bounds__(256)
void k_rope(const float* __restrict__ qkv,
            float* __restrict__ q_rope, float* __restrict__ k_new)
{
    const int QTOT = B * NH * QL * 64;   // 57344
    const int KTOT = B * NKV * QL * 64;  // 8192
    int idx = blockIdx.x * 256 + threadIdx.x;
    if (idx >= QTOT + KTOT) return;

    const bool isQ = idx < QTOT;
    const int t    = isQ ? idx : idx - QTOT;
    const int d    = t & 63;
    const int rest = t >> 6;
    const int s    = rest & 3;
    const int hh   = rest >> 2;          // b*NH+h  or  b*NKV+kh

    // inv_freq[d] = theta^(-2d/128) = theta^(-d/64)
    const float ang = (float)(PAST + s) * __powf(1000000.0f, -(float)d * (1.0f / 64.0f));
    const float cs = cosf(ang), sn = sinf(ang);

    if (isQ) {
        const int h = hh % NH, b = hh / NH;
        const float* src = qkv + (size_t)(b * QL + s) * NTOT + h * HD;
        const float x1 = src[d], x2 = src[d + 64];
        float* dst = q_rope + (size_t)hh * QL * HD + s * HD;
        dst[d]      = x1 * cs - x2 * sn;
        dst[d + 64] = x2 * cs + x1 * sn;
    } else {
        const int kh = hh % NKV, b = hh / NKV;
        const float* src = qkv + (size_t)(b * QL + s) * NTOT + NQ + kh * HD;
        const float x1 = src[d], x2 = src[d + 64];
        float* dst = k_new + (size_t)hh * QL * HD + s * HD;
        dst[d]      = x1 * cs - x2 * sn;
        dst[d + 64] = x2 * cs + x1 * sn;
    }
}

// ---------------------------------------------------------------------------
// Kernel 3: scores[b,h,s,j] = scale * q.k  (one thread per KV row), mask -1e9
// q tile (2 KB) staged Global->LDS with GLOBAL_LOAD_ASYNC_TO_LDS_B128
// (ASYNCcnt path), then s_wait_asynccnt + barrier before use.
// qs[] is the only LDS object in this kernel => workgroup LDS offset 0, so
// the LDS destination VGPR is simply tid*16 (HW adds the wave's LDS base).
// ---------------------------------------------------------------------------
__global__ __launch_bounds__(256)
void k_scores(const float* __restrict__ past_k, const float* __restrict__ q_rope,
              const float* __restrict__ k_new, float* __restrict__ scores)
{
    __shared__ __align__(16) float qs[QL * HD];   // at LDS offset 0
    const int bh = blockIdx.y;                    // 0..223

    if (threadIdx.x < (QL * HD) / 4) {            // 128 lanes x 16B = 2048B
        const float* gsrc = q_rope + (size_t)bh * QL * HD + threadIdx.x * 4;
        unsigned lds_off = threadIdx.x * 16;      // byte offset into qs[]
        asm volatile("global_load_async_to_lds_b128 %0, %1, off"
                     :: "v"(lds_off), "v"(gsrc) : "memory");
    }
    asm volatile("s_wait_asynccnt 0x0" ::: "memory");
    __syncthreads();

    const int j = blockIdx.x * 256 + threadIdx.x;
    if (j >= KVL) return;
    const int b = bh / NH, h = bh % NH;

    const float* krow;
    if (j < PAST)
        krow = past_k + ((size_t)(b * NH + h) * PAST + j) * HD;
    else
        krow = k_new + ((size_t)(b * NKV + h / NREP) * QL + (j - PAST)) * HD;

    float acc[QL] = {0.f, 0.f, 0.f, 0.f};
    for (int i = 0; i < HD / 4; ++i) {
        const float4 kv = reinterpret_cast<const float4*>(krow)[i];
#pragma unroll
        for (int s = 0; s < QL; ++s) {
            const float4 q = reinterpret_cast<const float4*>(qs + s * HD)[i];
            acc[s] += kv.x * q.x + kv.y * q.y + kv.z * q.z + kv.w * q.w;
        }
    }
#pragma unroll
    for (int s = 0; s < QL; ++s) {
        float v = acc[s] * SCALE;
        if (j > PAST + s) v = -1e9f;          // causal mask, same as reference
        scores[((size_t)bh * QL + s) * KVL + j] = v;
    }
}

// ---------------------------------------------------------------------------
// Kernel 4: row softmax over KVL, in place (scores -> probs)
// ---------------------------------------------------------------------------
__global__ __launch_bounds__(256)
void k_softmax(float* __restrict__ scores)
{
    __shared__ float red[256];
    float* x = scores + (size_t)blockIdx.x * KVL;
    const int t = threadIdx.x;

    float m = -3.4e38f;
    for (int j = t; j < KVL; j += 256) m = fmaxf(m, x[j]);
    red[t] = m; __syncthreads();
    for (int off = 128; off > 0; off >>= 1) {
        if (t < off) red[t] = fmaxf(red[t], red[t + off]);
        __syncthreads();
    }
    m = red[0]; __syncthreads();

    float ssum = 0.f;
    for (int j = t; j < KVL; j += 256) {
        const float e = __expf(x[j] - m);
        x[j] = e;
        ssum += e;
    }
    red[t] = ssum; __syncthreads();
    for (int off = 128; off > 0; off >>= 1) {
        if (t < off) red[t] += red[t + off];
        __syncthreads();
    }
    const float inv = 1.0f / red[0];
    for (int j = t; j < KVL; j += 256) x[j] *= inv;
}

// ---------------------------------------------------------------------------
// Kernel 5: zero attention accumulator (ws not re-zeroed between replays)
// ---------------------------------------------------------------------------
__global__ void k_zero(float* __restrict__ p, int n)
{
    const int i = blockIdx.x * 256 + threadIdx.x;
    if (i < n) p[i] = 0.f;
}

// ---------------------------------------------------------------------------
// Kernel 6: attn[b,h,s,:] += probs . V over a KV split; LDS + atomic reduce
// ---------------------------------------------------------------------------
__global__ __launch_bounds__(256)
void k_pv(const float* __restrict__ past_v, const float* __restrict__ qkv,
          const float* __restrict__ probs, float* __restrict__ attn)
{
    __shared__ float accs[QL * HD];
    const int bh = blockIdx.x;
    const int b = bh / NH, h = bh % NH;
    for (int i = threadIdx.x; i < QL * HD; i += 256) accs[i] = 0.f;
    __syncthreads();

    const int d4 = (threadIdx.x & 31) * 4;   // head-dim quad owned by this lane
    const int g  = threadIdx.x >> 5;         // row group 0..7
    const int CH = (KVL + 3) / 4;            // 1025
    const int start = blockIdx.y * CH;
    const int end   = min(KVL, start + CH);

    float ax[QL], ay[QL], az[QL], aw[QL];
#pragma unroll
    for (int s = 0; s < QL; ++s) { ax[s] = ay[s] = az[s] = aw[s] = 0.f; }

    const float* prow = probs + (size_t)bh * QL * KVL;
    for (int r = start + g; r < end; r += 8) {
        const float* vrow = (r < PAST)
            ? past_v + ((size_t)(b * NH + h) * PAST + r) * HD
            : qkv + (size_t)(b * QL + (r - PAST)) * NTOT + NQ + NK + (h / NREP) * HD;
        const float4 v = *reinterpret_cast<const float4*>(vrow + d4);
#pragma unroll
        for (int s = 0; s < QL; ++s) {
            const float p = prow[(size_t)s * KVL + r];
            ax[s] += p * v.x; ay[s] += p * v.y; az[s] += p * v.z; aw[s] += p * v.w;
        }
    }
#pragma unroll
    for (int s = 0; s < QL; ++s) {
        atomicAdd(&accs[s * HD + d4 + 0], ax[s]);
        atomicAdd(&accs[s * HD + d4 + 1], ay[s]);
        atomicAdd(&accs[s * HD + d4 + 2], az[s]);
        atomicAdd(&accs[s * HD + d4 + 3], aw[s]);
    }
    __syncthreads();
    for (int i = threadIdx.x; i < QL * HD; i += 256)
        atomicAdd(&attn[(size_t)bh * QL * HD + i], accs[i]);
}

// ---------------------------------------------------------------------------
// Kernel 7: output projection out[32][3584] = A(gathered) @ Wo, fp32 WMMA
// A[row][k]: row=b*4+s, k=h*128+d -> attn[((b*NH+h)*QL+s)*HD + d]
// ---------------------------------------------------------------------------
__global__ __launch_bounds__(32)
void k_out_gemm(const float* __restrict__ attn, const float* __restrict__ Wo,
                float* __restrict__ out)
{
    const int lane   = threadIdx.x;
    const int n0     = blockIdx.x * 16;
    const int tile_m = blockIdx.y;
    const int half = lane >> 4;
    const int l    = lane & 15;

    const int arow = tile_m * 16 + l;            // 0..31
    const int ab = arow >> 2, as = arow & 3;
    const float* Abase = attn + (size_t)ab * NH * QL * HD + as * HD;
    const float* Bcol  = Wo + (n0 + l);

    v8f c = {};
    for (int k0 = 0; k0 < NQ; k0 += 4) {
        const int ka = k0 + half * 2;
        const int h0 = ka >> 7,        d0 = ka & 127;
        const int h1 = (ka + 1) >> 7,  d1 = (ka + 1) & 127;
        v2f a, b;
        a.x = Abase[(size_t)h0 * QL * HD + d0];
        a.y = Abase[(size_t)h1 * QL * HD + d1];
        b.x = Bcol[(size_t)ka * HID];
        b.y = Bcol[(size_t)(ka + 1) * HID];
        c = __builtin_amdgcn_wmma_f32_16x16x4_f32(false, a, false, b,
                                                  (short)0, c, false, false);
    }
    const int mb = tile_m * 16 + half * 8;
#pragma unroll
    for (int r = 0; r < 8; ++r)
        out[(size_t)(mb + r) * HID + n0 + l] = c[r];
}

// ---------------------------------------------------------------------------
extern "C" void kernel_launch(void* const* d_in, const int* in_sizes, int n_in,
                              void* d_out, int out_size, void* d_ws, size_t ws_size,
                              hipStream_t stream)
{
    const float* hidden = (const float*)d_in[0];
    const float* past_k = (const float*)d_in[1];
    const float* past_v = (const float*)d_in[2];
    // d_in[3] = attention_mask (pure causal; recomputed analytically)
    const float* Wq = (const float*)d_in[4];
    const float* bq = (const float*)d_in[5];
    const float* Wk = (const float*)d_in[6];
    const float* bk = (const float*)d_in[7];
    const float* Wv = (const float*)d_in[8];
    const float* bv = (const float*)d_in[9];
    const float* Wo = (const float*)d_in[10];

    float* ws     = (float*)d_ws;
    float* qkv    = ws;                               // 32*4608      = 147456
    float* q_rope = qkv    + (size_t)ROWS * NTOT;     // 8*28*4*128   = 114688
    float* k_new  = q_rope + (size_t)B * NH * QL * HD;  // 8*4*4*128  = 16384
    float* attn   = k_new  + (size_t)B * NKV * QL * HD; // 8*28*4*128 = 114688
    float* scores = attn   + (size_t)B * NH * QL * HD;  // 8*28*4*4100

    // 1) QKV projection (WMMA fp32)
    dim3 g1(NTOT / 16, ROWS / 16);
    k_qkv_gemm<<<g1, 32, 0, stream>>>(hidden, Wq, bq, Wk, bk, Wv, bv, qkv);

    // 2) RoPE
    k_rope<<<(B * NH * QL * 64 + B * NKV * QL * 64 + 255) / 256, 256, 0, stream>>>(
        qkv, q_rope, k_new);

    // 3) scores (async Global->LDS q staging)
    dim3 g3((KVL + 255) / 256, B * NH);
    k_scores<<<g3, 256, 0, stream>>>(past_k, q_rope, k_new, scores);

    // 4) softmax
    k_softmax<<<B * NH * QL, 256, 0, stream>>>(scores);

    // 5) zero attn accumulator
    const int attn_n = B * NH * QL * HD;
    k_zero<<<(attn_n + 255) / 256, 256, 0, stream>>>(attn, attn_n);

    // 6) probs @ V with KV-split
    dim3 g6(B * NH, 4);
    k_pv<<<g6, 256, 0, stream>>>(past_v, qkv, scores, attn);

    // 7) output projection (WMMA fp32)
    dim3 g7(HID / 16, ROWS / 16);
    k_out_gemm<<<g7, 32, 0, stream>>>(attn, Wo, (float*)d_out);
}